// CTCLayer_77068893159464
// MI455X (gfx1250) — compile-verified
//
#include <hip/hip_runtime.h>
#include <cstdint>

#define EPSV (1e-7f)

// CDNA5 async global->LDS gather (gfx1250). VDST = per-lane LDS byte address,
// VADDR = per-lane 64-bit global address, saddr = off (GV mode). Tracked by ASYNCcnt.
#define ASYNC_G2L_B32(GP, LP)                                                   \
  asm volatile("global_load_async_to_lds_b32 %0, %1, off"                       \
               :: "v"((unsigned)(uintptr_t)(LP)),                               \
                  "v"((unsigned long long)(uintptr_t)(GP))                      \
               : "memory")

#define WAIT_ASYNCCNT_9() asm volatile("s_wait_asynccnt 9" ::: "memory")
#define WAIT_ASYNCCNT_0() asm volatile("s_wait_asynccnt 0" ::: "memory")

// One wave32 per batch element. Lane l owns CTC states s = 4l..4l+3
// (lane 31 additionally owns s = 128). Alpha kept in registers in the
// PROBABILITY domain with wave-wide rescaling every 4 steps (Rabiner scaling):
//   new[s] = (p[ext[s]]+eps) * (a[s] + a[s-1] + skip*a[s-2])
// which is the exact CTC recursion with all transcendentals hoisted out of the
// critical sequential chain. Per-step probability gather (2 labels + blank per
// lane) is software-pipelined D=4 deep through LDS with async loads.
__global__ __launch_bounds__(32)
void ctc_loss_kernel(const int* __restrict__ y_true,     // [B, 64]
                     const float* __restrict__ y_pred,   // [B, T, 128]
                     const int* __restrict__ in_len,     // [B, 1]
                     const int* __restrict__ lab_len,    // [B, 1]
                     float* __restrict__ out,            // [B]
                     int T) {
  constexpr int K = 128;
  constexpr int L = 64;
  constexpr int D = 4;  // pipeline depth == rescale cadence

  const int b = blockIdx.x;
  const int l = threadIdx.x;  // 0..31

  // D stages; each stage: 32 lanes * 4 floats (p_lab0, p_lab1, p_blank, pad)
  __shared__ __align__(16) float lpstage[D * 128];
  __shared__ __align__(16) float aLDS[132];

  const float* __restrict__ base = y_pred + (size_t)b * (size_t)T * (size_t)K;

  // Labels 2l, 2l+1 for this lane.
  const int2 lab2 = ((const int2*)(y_true + (size_t)b * L))[l];
  const int lab0 = lab2.x;
  const int lab1 = lab2.y;
  const int labPrev = __shfl_up(lab1, 1, 32);  // label_{2l-1}
  // skip masks as float multipliers (even/blank states never skip)
  const float sk0 = ((l != 0) && (lab0 != labPrev)) ? 1.0f : 0.0f;  // s=4l+1 (j=2l)
  const float sk1 = (lab1 != lab0) ? 1.0f : 0.0f;                   // s=4l+3 (j=2l+1)

  const int len = in_len[b];

  // alpha init (prob domain): alpha[0] = 1, rest = 0
  float a0 = (l == 0) ? 1.0f : 0.0f;  // s = 4l
  float a1 = 0.0f;                    // s = 4l+1
  float a2 = 0.0f;                    // s = 4l+2
  float a3 = 0.0f;                    // s = 4l+3
  float a4 = 0.0f;                    // s = 128 (lane 31 only)
  float ls = 0.0f;                    // accumulated log-scale (wave uniform)

  // Prologue: fill the async pipeline and warm L2 a bit further ahead.
  const int pre = (len < D) ? len : D;
  for (int t = 0; t < pre; ++t) {
    const float* row = base + (size_t)t * K;
    float* dst = &lpstage[t * 128 + (l << 2)];
    ASYNC_G2L_B32(row + lab0, dst);
    ASYNC_G2L_B32(row + lab1, dst + 1);
    ASYNC_G2L_B32(row + (K - 1), dst + 2);
  }
  for (int r = D; r < 16; ++r)
    __builtin_prefetch(base + (size_t)r * K + (l << 2), 0, 2);

  int t = 0;
  // Steady state: t is always a multiple of D, so ring slot == unroll index
  // (constant LDS offsets). Sub-step waits to ASYNCcnt<=9 = 3*(D-1), which
  // guarantees the oldest stage completed (async completions are in-order).
  for (; t + 2 * D <= len; t += D) {
#pragma unroll
    for (int i = 0; i < D; ++i) {
      WAIT_ASYNCCNT_9();
      const float* sp = &lpstage[i * 128 + (l << 2)];
      const float f0 = sp[0] + EPSV;
      const float f1 = sp[1] + EPSV;
      const float fb = sp[2] + EPSV;

      float am1 = __shfl_up(a3, 1, 32);  // alpha[4l-1]
      if (l == 0) am1 = 0.0f;

      const float n0 = fb * (a0 + am1);              // s=4l   (blank)
      const float n1 = f0 * (a1 + a0 + sk0 * am1);   // s=4l+1 (label 2l)
      const float n2 = fb * (a2 + a1);               // s=4l+2 (blank)
      const float n3 = f1 * (a3 + a2 + sk1 * a1);    // s=4l+3 (label 2l+1)
      const float n4 = fb * (a4 + a3);               // s=128  (lane 31)
      a0 = n0; a1 = n1; a2 = n2; a3 = n3; a4 = n4;

      // Refill slot i for step t+i+D (guaranteed < len by loop bound).
      const int tn = t + i + D;
      const float* row = base + (size_t)tn * K;
      float* dst = &lpstage[i * 128 + (l << 2)];
      ASYNC_G2L_B32(row + lab0, dst);
      ASYNC_G2L_B32(row + lab1, dst + 1);
      ASYNC_G2L_B32(row + (K - 1), dst + 2);
      // Speculative L2 warm-up ~12 rows past the async window (dropped on fault).
      __builtin_prefetch(base + (size_t)(tn + 12) * K + (l << 2), 0, 2);
    }
    // Rescale: wave-wide max, scale to ~1, accumulate log of the scale.
    float m = fmaxf(fmaxf(fmaxf(a0, a1), fmaxf(a2, a3)), a4);
#pragma unroll
    for (int off = 16; off; off >>= 1) m = fmaxf(m, __shfl_xor(m, off, 32));
    const float r = __builtin_amdgcn_rcpf(m);
    ls += __logf(m);
    a0 *= r; a1 *= r; a2 *= r; a3 *= r; a4 *= r;
  }

  // Tail (<= 2*D-1 steps): fully drain before each consume; dynamic ring slot.
  for (; t < len; ++t) {
    WAIT_ASYNCCNT_0();
    const float* sp = &lpstage[(t & (D - 1)) * 128 + (l << 2)];
    const float f0 = sp[0] + EPSV;
    const float f1 = sp[1] + EPSV;
    const float fb = sp[2] + EPSV;

    float am1 = __shfl_up(a3, 1, 32);
    if (l == 0) am1 = 0.0f;

    const float n0 = fb * (a0 + am1);
    const float n1 = f0 * (a1 + a0 + sk0 * am1);
    const float n2 = fb * (a2 + a1);
    const float n3 = f1 * (a3 + a2 + sk1 * a1);
    const float n4 = fb * (a4 + a3);
    a0 = n0; a1 = n1; a2 = n2; a3 = n3; a4 = n4;

    const int tn = t + D;
    if (tn < len) {
      const float* row = base + (size_t)tn * K;
      float* dst = &lpstage[(tn & (D - 1)) * 128 + (l << 2)];
      ASYNC_G2L_B32(row + lab0, dst);
      ASYNC_G2L_B32(row + lab1, dst + 1);
      ASYNC_G2L_B32(row + (K - 1), dst + 2);
    }
  }
  WAIT_ASYNCCNT_0();

  // Spill alpha to LDS for the final indexed reads (same wave: DS is in-order,
  // no barrier needed; wave_barrier only pins compiler ordering).
  aLDS[(l << 2) + 0] = a0;
  aLDS[(l << 2) + 1] = a1;
  aLDS[(l << 2) + 2] = a2;
  aLDS[(l << 2) + 3] = a3;
  if (l == 31) aLDS[128] = a4;
  __builtin_amdgcn_wave_barrier();

  if (l == 0) {
    const int LL = lab_len[b];
    int i1 = 2 * LL;
    if (i1 > 128) i1 = 128;
    int i2 = i1 - 1;
    if (i2 < 0) i2 = 0;
    const float e = aLDS[i1] + aLDS[i2];
    // loss = -logaddexp(log a[i1]+ls, log a[i2]+ls) = -(log(a[i1]+a[i2]) + ls)
    out[b] = -(__logf(e) + ls);
  }
}

extern "C" void kernel_launch(void* const* d_in, const int* in_sizes, int n_in,
                              void* d_out, int out_size, void* d_ws, size_t ws_size,
                              hipStream_t stream) {
  // setup_inputs order: y_true(int32), y_pred(f32), input_length(int32), label_length(int32)
  const int* y_true = (const int*)d_in[0];
  const float* y_pred = (const float*)d_in[1];
  const int* in_len = (const int*)d_in[2];
  const int* lab_len = (const int*)d_in[3];
  float* out = (float*)d_out;

  const int B = in_sizes[2];                 // 512
  const int T = in_sizes[1] / (B * 128);     // 1024

  ctc_loss_kernel<<<dim3(B), dim3(32), 0, stream>>>(y_true, y_pred, in_len,
                                                    lab_len, out, T);
}